// FrequencyAnalysis_40922448396797
// MI455X (gfx1250) — compile-verified
//
#include <hip/hip_runtime.h>
#include <hip/hip_bf16.h>

typedef __attribute__((ext_vector_type(2))) float v2f;
typedef __attribute__((ext_vector_type(8))) float v8f;

#define PATCH_ELEMS 3072   // C*H*W = 3*32*32
#define NPATCH      256    // N
#define NBATCH      64     // B
#define KTOP        9

// ---------------------------------------------------------------------------
// Kernel 1: per-patch abs-sum via V_WMMA_F32_16X16X4_F32.
// Each wave (32 lanes) owns 16 consecutive patches. A = |x| (16 patches x 4 k),
// B = ones (4x16), so D[m,n] = rowsum_m + C[m,n] accumulates the exact f32
// abs-sum of each patch in every column of row m. 3072/4 = 768 WMMAs per wave.
// ---------------------------------------------------------------------------
__global__ __launch_bounds__(256) void fa_grade_kernel(
    const float* __restrict__ x, float* __restrict__ grades) {
  const int waveInBlk = threadIdx.x >> 5;
  const int lane      = threadIdx.x & 31;
  const int waveId    = blockIdx.x * 8 + waveInBlk;   // 0..1023
  const int m         = lane & 15;                    // A-matrix row (patch)
  const int koff      = (lane >> 4) << 1;             // lanes 16-31 carry K+2,K+3
  const size_t patch0 = (size_t)waveId * 16;          // 16 patches per wave

  const float* base = x + (patch0 + (size_t)m) * PATCH_ELEMS + koff;

  v2f ones; ones.x = 1.0f; ones.y = 1.0f;
  v8f c = {};  // f32 accumulator, starts at 0

  for (int k = 0; k < PATCH_ELEMS; k += 16) {
    __builtin_prefetch(base + k + 256, 0, 1);  // global_prefetch_b8, ~1KB ahead
#pragma unroll
    for (int s = 0; s < 4; ++s) {
      const float* p = base + k + s * 4;
      v2f a;
      a.x = __builtin_fabsf(p[0]);
      a.y = __builtin_fabsf(p[1]);
      // 8 args: (neg_a, A, neg_b, B, c_mod, C, reuse_a, reuse_b)
      c = __builtin_amdgcn_wmma_f32_16x16x4_f32(
          false, a, false, ones, (short)0, c, false, false);
    }
  }

  // D layout (32-bit 16x16): VGPR v, lanes 0-15 -> M=v ; lanes 16-31 -> M=v+8.
  // Lane 0 holds column 0 of rows 0..7, lane 16 holds rows 8..15.
  if (m == 0) {
    const int mbase = (lane >> 4) * 8;
#pragma unroll
    for (int v = 0; v < 8; ++v)
      grades[patch0 + mbase + v] = c[v];
  }
}

// ---------------------------------------------------------------------------
// Kernel 2: per-batch top-9 (max) and bottom-9 (min) argselect over 256 grades.
// Tie-break: lower index wins (matches jax.lax.top_k on grades / -grades).
// ---------------------------------------------------------------------------
__global__ __launch_bounds__(256) void fa_topk_kernel(
    const float* __restrict__ grades,
    int* __restrict__ hi_idx, int* __restrict__ lo_idx) {
  __shared__ float sval[NPATCH];
  __shared__ float wval[NPATCH];
  __shared__ int   widx[NPATCH];
  const int b = blockIdx.x;
  const int t = threadIdx.x;

  // ---- top 9 (argmax passes) ----
  sval[t] = grades[b * NPATCH + t];
  __syncthreads();
  for (int pass = 0; pass < KTOP; ++pass) {
    wval[t] = sval[t]; widx[t] = t;
    __syncthreads();
    for (int off = 128; off > 0; off >>= 1) {
      if (t < off) {
        float v1 = wval[t], v2 = wval[t + off];
        int   i1 = widx[t], i2 = widx[t + off];
        if ((v2 > v1) || (v2 == v1 && i2 < i1)) { wval[t] = v2; widx[t] = i2; }
      }
      __syncthreads();
    }
    if (t == 0) {
      int arg = widx[0];
      hi_idx[b * KTOP + pass] = arg;
      sval[arg] = -__builtin_inff();
    }
    __syncthreads();
  }

  // ---- bottom 9 (argmin passes) ----
  sval[t] = grades[b * NPATCH + t];
  __syncthreads();
  for (int pass = 0; pass < KTOP; ++pass) {
    wval[t] = sval[t]; widx[t] = t;
    __syncthreads();
    for (int off = 128; off > 0; off >>= 1) {
      if (t < off) {
        float v1 = wval[t], v2 = wval[t + off];
        int   i1 = widx[t], i2 = widx[t + off];
        if ((v2 < v1) || (v2 == v1 && i2 < i1)) { wval[t] = v2; widx[t] = i2; }
      }
      __syncthreads();
    }
    if (t == 0) {
      int arg = widx[0];
      lo_idx[b * KTOP + pass] = arg;
      sval[arg] = __builtin_inff();
    }
    __syncthreads();
  }
}

// ---------------------------------------------------------------------------
// Kernel 3: gather selected patches. One block per (b, slot) copies one 12KB
// patch with float4 transactions. Output = [highest 64x9x3072 | lowest ...].
// ---------------------------------------------------------------------------
__global__ __launch_bounds__(256) void fa_gather_kernel(
    const float* __restrict__ x,
    const int* __restrict__ hi_idx, const int* __restrict__ lo_idx,
    float* __restrict__ out) {
  const int blk = blockIdx.x;          // 64 * 18
  const int b   = blk / (2 * KTOP);
  const int j   = blk % (2 * KTOP);

  int src;
  size_t dstOff;
  if (j < KTOP) {
    src    = hi_idx[b * KTOP + j];
    dstOff = ((size_t)b * KTOP + j) * PATCH_ELEMS;
  } else {
    const int jj = j - KTOP;
    src    = lo_idx[b * KTOP + jj];
    dstOff = (size_t)NBATCH * KTOP * PATCH_ELEMS +
             ((size_t)b * KTOP + jj) * PATCH_ELEMS;
  }

  const float4* s = (const float4*)(x + ((size_t)b * NPATCH + src) * PATCH_ELEMS);
  float4*       d = (float4*)(out + dstOff);
#pragma unroll
  for (int i = 0; i < 3; ++i)                 // 768 float4 / 256 threads
    d[threadIdx.x + i * 256] = s[threadIdx.x + i * 256];
}

extern "C" void kernel_launch(void* const* d_in, const int* in_sizes, int n_in,
                              void* d_out, int out_size, void* d_ws, size_t ws_size,
                              hipStream_t stream) {
  const float* x = (const float*)d_in[0];
  // d_in[1]/d_in[2] are k_highest/k_lowest device scalars; fixed at 9 by setup.

  float* grades = (float*)d_ws;                                    // 16384 f32
  int*   hi_idx = (int*)((char*)d_ws + NBATCH * NPATCH * sizeof(float));
  int*   lo_idx = hi_idx + NBATCH * KTOP;

  // 1024 waves total: 128 blocks x 8 waves, 16 patches each.
  fa_grade_kernel<<<128, 256, 0, stream>>>(x, grades);
  fa_topk_kernel<<<NBATCH, 256, 0, stream>>>(grades, hi_idx, lo_idx);
  fa_gather_kernel<<<NBATCH * 2 * KTOP, 256, 0, stream>>>(x, hi_idx, lo_idx,
                                                          (float*)d_out);
}